// Get_Joints_2448131359333
// MI455X (gfx1250) — compile-verified
//
#include <hip/hip_runtime.h>
#include <math.h>

typedef float v2f __attribute__((ext_vector_type(2)));
typedef float v8f __attribute__((ext_vector_type(8)));

#define NUM_VERTS  6890
#define NUM_JOINTS 24
#define NUM_BETAS  10
#define NCOLS      33   // cols: 0..2 = v_template (J_base), 3 + k*10 + l = shapedirs[:,k,l]

// ---------------------------------------------------------------------------
// Kernel 1: JW[24][33] = J_regressor[24][6890] x V[6890][33]  via f32 WMMA.
//   V[i][c] = (c < 3) ? v_template[i][c] : shapedirs[i*30 + (c-3)]
//
// Branchless main loop: tiles OVERLAP so every lane's load address is valid.
//   M tiles: row base {0, 8}  -> rows 0..15 / 8..23 (store rows >= {0,16})
//   N tiles: col base {0,16,17} -> cols 0..15 / 16..31 / 17..32
//                                  (store cols >= {0,16,32})
// K loop: 1722 unguarded chunks of 4, plus one guarded tail chunk (6890%4==2).
//
// WMMA f32 16x16x4 operand layout (ISA 7.12.2):
//   A: lane L -> M = L&15;  VGPR v -> K = v + 2*(L>>4)
//   B: lane L -> N = L&15;  VGPR v -> K = v + 2*(L>>4)
//   C/D: lane L, VGPR v -> M = v + 8*(L>>4), N = L&15
// ---------------------------------------------------------------------------
__global__ __launch_bounds__(32)
void jw_gemm_wmma(const float* __restrict__ JR,   // [24][6890]
                  const float* __restrict__ vt,   // [6890][3]
                  const float* __restrict__ sd,   // [6890][3][10]
                  float* __restrict__ JW)         // [24][33]
{
    constexpr int mBase[2] = {0, 8};
    constexpr int mMin [2] = {0, 16};
    constexpr int nBase[3] = {0, 16, 17};
    constexpr int nMin [3] = {0, 16, 32};

    const int lane = threadIdx.x & 31;
    const int l15  = lane & 15;
    const int half = lane >> 4;          // 0 | 1
    const int kOff = half * 2;           // K offset of this half-wave

    const int mRow = mBase[blockIdx.x] + l15;   // always < 24
    const int nCol = nBase[blockIdx.y] + l15;   // always < 33

    // A: two K-consecutive elements per lane -> one aligned b64 load.
    //    (row*6890 and kOff are even, so 'ap' stays 8-byte aligned)
    const float* ap = JR + (size_t)mRow * NUM_VERTS + kOff;

    // B: column nCol of the virtual V matrix; stride 3 (template) or 30 (dirs).
    const int bStride = (nCol < 3) ? 3 : 30;
    const float* bBase = (nCol < 3) ? (vt + nCol) : (sd + (nCol - 3));
    const float* bp0 = bBase + (size_t)kOff * bStride;
    const float* bp1 = bp0 + bStride;
    const size_t bStep = (size_t)4 * bStride;

    v8f acc = {};
    #pragma unroll 4
    for (int it = 0; it < 1722; ++it) {      // K = [0, 6888)
        v2f a = *(const v2f*)ap;             // A[mRow][k], A[mRow][k+1]
        v2f b;
        b.x = *bp0;                          // V[k  ][nCol]
        b.y = *bp1;                          // V[k+1][nCol]
        acc = __builtin_amdgcn_wmma_f32_16x16x4_f32(false, a, false, b,
                                                    (short)0, acc, false, false);
        ap  += 4;
        bp0 += bStep;
        bp1 += bStep;
    }
    {   // tail: K = 6888..6891; only 6888/6889 exist (half-wave 0's lanes)
        v2f a = {0.0f, 0.0f};
        v2f b = {0.0f, 0.0f};
        if (half == 0) {
            a = *(const v2f*)ap;
            b.x = *bp0;
            b.y = *bp1;
        }
        acc = __builtin_amdgcn_wmma_f32_16x16x4_f32(false, a, false, b,
                                                    (short)0, acc, false, false);
    }

    const int rowLo = mMin[blockIdx.x];
    const int colLo = nMin[blockIdx.y];
    #pragma unroll
    for (int v = 0; v < 8; ++v) {
        const int row = mBase[blockIdx.x] + v + 8 * half;
        const int col = nCol;
        if (row >= rowLo && col >= colLo)
            JW[row * NCOLS + col] = acc[v];
    }
}

// ---------------------------------------------------------------------------
// Kernel 2: per-sample Rodrigues + kinematic chain. One wave32 block per batch
// element; lane j (< 24) owns joint j. Tree resolved level-by-level via LDS.
// ---------------------------------------------------------------------------
__global__ __launch_bounds__(32)
void smpl_chain(const float* __restrict__ pose,   // [B][72]
                const float* __restrict__ trans,  // [B][3]
                const float* __restrict__ betas,  // [B][10]
                const float* __restrict__ JW,     // [24][33]
                float* __restrict__ out)          // [B][24][3]
{
    constexpr int parent[NUM_JOINTS] =
        {-1,0,0,0,1,2,3,4,5,6,7,8,9,9,9,12,13,14,16,17,18,19,20,21};
    constexpr int depth[NUM_JOINTS] =
        { 0,1,1,1,2,2,2,3,3,3,4,4,4,4,4, 5, 5, 5, 6, 6, 7, 7, 8, 8};

    __shared__ float sJ[NUM_JOINTS][3];   // shaped joint positions
    __shared__ float sC[NUM_JOINTS][12];  // chain transforms, 3 rows of [R|t]

    const int b = blockIdx.x;
    const int j = threadIdx.x;

    float R[9]    = {};
    float relT[3] = {};

    if (j < NUM_JOINTS) {
        // J[b][j] = JW[j][0:3] + sum_l betas[b][l] * JW[j][3 + k*10 + l]
        float bt[NUM_BETAS];
        #pragma unroll
        for (int l = 0; l < NUM_BETAS; ++l) bt[l] = betas[b * NUM_BETAS + l];
        const float* jw = JW + j * NCOLS;
        #pragma unroll
        for (int k = 0; k < 3; ++k) {
            float s = jw[k];
            #pragma unroll
            for (int l = 0; l < NUM_BETAS; ++l) s += bt[l] * jw[3 + k * 10 + l];
            sJ[j][k] = s;
        }
        // Rodrigues (reference numerics: angle = |p + eps|, dir = p / angle)
        const float px = pose[b * 72 + 3 * j + 0];
        const float py = pose[b * 72 + 3 * j + 1];
        const float pz = pose[b * 72 + 3 * j + 2];
        const float ex = px + 1e-8f, ey = py + 1e-8f, ez = pz + 1e-8f;
        const float ang = sqrtf(ex * ex + ey * ey + ez * ez);
        const float inv = 1.0f / ang;
        const float rx = px * inv, ry = py * inv, rz = pz * inv;
        const float sn = sinf(ang);
        const float om = 1.0f - cosf(ang);
        R[0] = 1.0f + om * (-(ry * ry + rz * rz));
        R[1] = -sn * rz + om * (rx * ry);
        R[2] =  sn * ry + om * (rx * rz);
        R[3] =  sn * rz + om * (rx * ry);
        R[4] = 1.0f + om * (-(rx * rx + rz * rz));
        R[5] = -sn * rx + om * (ry * rz);
        R[6] = -sn * ry + om * (rx * rz);
        R[7] =  sn * rx + om * (ry * rz);
        R[8] = 1.0f + om * (-(rx * rx + ry * ry));
    }
    __syncthreads();

    if (j < NUM_JOINTS) {
        const int p = parent[j];
        #pragma unroll
        for (int k = 0; k < 3; ++k)
            relT[k] = (p < 0) ? sJ[j][k] : (sJ[j][k] - sJ[p][k]);
    }
    if (j == 0) {  // root transform
        sC[0][0] = R[0]; sC[0][1] = R[1]; sC[0][2]  = R[2]; sC[0][3]  = relT[0];
        sC[0][4] = R[3]; sC[0][5] = R[4]; sC[0][6]  = R[5]; sC[0][7]  = relT[1];
        sC[0][8] = R[6]; sC[0][9] = R[7]; sC[0][10] = R[8]; sC[0][11] = relT[2];
    }
    __syncthreads();

    for (int lv = 1; lv <= 8; ++lv) {
        if (j < NUM_JOINTS && depth[j] == lv) {
            const int p = parent[j];
            float P[12];
            #pragma unroll
            for (int t = 0; t < 12; ++t) P[t] = sC[p][t];
            float C[12];
            #pragma unroll
            for (int r = 0; r < 3; ++r) {
                const float p0 = P[r * 4 + 0], p1 = P[r * 4 + 1], p2 = P[r * 4 + 2];
                #pragma unroll
                for (int c = 0; c < 3; ++c)
                    C[r * 4 + c] = p0 * R[c] + p1 * R[3 + c] + p2 * R[6 + c];
                C[r * 4 + 3] = p0 * relT[0] + p1 * relT[1] + p2 * relT[2] + P[r * 4 + 3];
            }
            #pragma unroll
            for (int t = 0; t < 12; ++t) sC[j][t] = C[t];
        }
        __syncthreads();
    }

    if (j < NUM_JOINTS) {
        #pragma unroll
        for (int k = 0; k < 3; ++k)
            out[(b * NUM_JOINTS + j) * 3 + k] = sC[j][k * 4 + 3] + trans[b * 3 + k];
    }
}

// ---------------------------------------------------------------------------
extern "C" void kernel_launch(void* const* d_in, const int* in_sizes, int n_in,
                              void* d_out, int out_size, void* d_ws, size_t ws_size,
                              hipStream_t stream)
{
    // setup_inputs order: pose, trans, betas, v_template, shapedirs, J_regressor, parents
    const float* pose  = (const float*)d_in[0];
    const float* trans = (const float*)d_in[1];
    const float* betas = (const float*)d_in[2];
    const float* vt    = (const float*)d_in[3];
    const float* sd    = (const float*)d_in[4];
    const float* JRg   = (const float*)d_in[5];
    (void)n_in; (void)ws_size; (void)out_size;

    float* JW  = (float*)d_ws;           // 24*33 floats of scratch
    float* out = (float*)d_out;          // [B][24][3]
    const int batch = in_sizes[0] / (NUM_JOINTS * 3);

    dim3 g1(2, 3);                        // 2 M-tiles x 3 N-tiles (overlapping)
    jw_gemm_wmma<<<g1, 32, 0, stream>>>(JRg, vt, sd, JW);
    smpl_chain<<<batch, 32, 0, stream>>>(pose, trans, betas, JW, out);
}